// EfficientExpandingAttention_67284957659386
// MI455X (gfx1250) — compile-verified
//
#include <hip/hip_runtime.h>
#include <math.h>

#define T_LEN   8192
#define N_EMBD  2048
#define LDW     (2*N_EMBD + 1)   // 4097 columns in W
#define N_ITERS 30

typedef __attribute__((ext_vector_type(2))) float v2f;
typedef __attribute__((ext_vector_type(8))) float v8f;

// ---- workspace layout (float offsets) ----
#define WS_QLAST 0
#define WS_W2    (WS_QLAST + N_EMBD)        // 2048
#define WS_WV    (WS_W2   + N_EMBD)        // 4096
#define WS_E     (WS_WV   + N_EMBD)        // 6144
#define WS_V     (WS_E    + T_LEN)         // 14336
#define WS_SE    (WS_V    + T_LEN)         // 22528
#define WS_SW    (WS_SE   + T_LEN + 64)    // 30784
#define WS_SEV   (WS_SW   + T_LEN + 64)    // 39040
// end = 47296 floats = ~185 KB of d_ws

// ------------------------------------------------------------------
// Kernel 1: q_last[j] = sum_i x[T-1,i] * W[i,j]   (j = 0..2047)
// also stage wv[i] = W[i, 4096]
// ------------------------------------------------------------------
__global__ void k_qlast(const float* __restrict__ x,
                        const float* __restrict__ W,
                        float* __restrict__ ws) {
    int j = blockIdx.x * blockDim.x + threadIdx.x;
    if (j >= N_EMBD) return;
    const float* xlast = x + (size_t)(T_LEN - 1) * N_EMBD;
    float acc = 0.f;
    for (int i = 0; i < N_EMBD; ++i)
        acc = fmaf(xlast[i], W[(size_t)i * LDW + j], acc);
    ws[WS_QLAST + j] = acc;
    ws[WS_WV + j]    = W[(size_t)j * LDW + 2 * N_EMBD];   // v column
}

// ------------------------------------------------------------------
// Kernel 2: w2[i] = sum_j W[i, 2048+j] * q_last[j]  (wave per row)
// ------------------------------------------------------------------
__global__ void k_w2(const float* __restrict__ W, float* __restrict__ ws) {
    int wave = threadIdx.x >> 5;
    int lane = threadIdx.x & 31;
    int r = blockIdx.x * (blockDim.x >> 5) + wave;
    if (r >= N_EMBD) return;
    const float* q   = ws + WS_QLAST;
    const float* row = W + (size_t)r * LDW + N_EMBD;
    float acc = 0.f;
    for (int j = lane; j < N_EMBD; j += 32)
        acc = fmaf(row[j], q[j], acc);
    #pragma unroll
    for (int off = 16; off > 0; off >>= 1)
        acc += __shfl_down(acc, off, 32);
    if (lane == 0) ws[WS_W2 + r] = acc;
}

// ------------------------------------------------------------------
// Kernel 3: fused  eraw[t] = x[t]·w2 ,  v[t] = x[t]·wv   via
// V_WMMA_F32_16X16X4_F32.  One wave per 16-row tile of x.
//   A (16x4 f32): lane%16 = M row, VGPR pair = K {2h,2h+1}, h=lane/16
//   B (4x16 f32): lane%16 = N col, VGPR pair = K {2h,2h+1}
//     col 0 = w2, col 1 = wv; cols 2..15 carry don't-care (w2) data —
//     they only affect D columns 2..15 which are never read.
//   D (16x16 f32): lane%16 = N, M = vgpr + 8*(lane/16)
// Branchless hot loop: per-lane A/B base pointers hoisted, EXEC
// untouched (WMMA requires EXEC all-ones).
// ------------------------------------------------------------------
__global__ void k_ev(const float* __restrict__ x, float* __restrict__ ws) {
    int wave = threadIdx.x >> 5;
    int lane = threadIdx.x & 31;
    int tile = blockIdx.x * (blockDim.x >> 5) + wave;    // exactly 512 tiles
    int h = lane >> 4;        // lane half selects K pair / M half
    int n = lane & 15;        // M row for A, N col for B/D

    // lane 0 -> w2 (col 0), lane 1 -> wv (col 1), lanes >=2 -> w2 (ignored cols)
    const float* bbase = ws + ((n == 1) ? WS_WV : WS_W2);
    const float* xa = x + (size_t)(tile * 16 + n) * N_EMBD + 2 * h;
    const float* ba = bbase + 2 * h;

    v8f c = {};
    #pragma unroll 8
    for (int kb = 0; kb < N_EMBD; kb += 4) {
        v2f a = *(const v2f*)(xa + kb);      // global_load_b64
        v2f b = *(const v2f*)(ba + kb);      // global_load_b64 (L2-resident)
        c = __builtin_amdgcn_wmma_f32_16x16x4_f32(
                /*neg_a=*/false, a, /*neg_b=*/false, b,
                /*c_mod=*/(short)0, c, /*reuse_a=*/false, /*reuse_b=*/false);
    }

    const float scale = 0.01f / sqrtf((float)N_EMBD);
    if (n == 0) {            // column 0 of D = x·w2  ->  e
        #pragma unroll
        for (int mm = 0; mm < 8; ++mm) {
            int row = tile * 16 + h * 8 + mm;
            ws[WS_E + row] = (row == T_LEN - 1) ? 0.f : expf(c[mm] * scale);
        }
    } else if (n == 1) {     // column 1 of D = x·wv  ->  v_vec
        #pragma unroll
        for (int mm = 0; mm < 8; ++mm) {
            int row = tile * 16 + h * 8 + mm;
            ws[WS_V + row] = c[mm];
        }
    }
}

// ------------------------------------------------------------------
// Kernel 4: suffix sums of e, e*(T-1-t), e*v  then the 30-step scan.
// Single block of 256 threads, 32 elements per thread.
// ------------------------------------------------------------------
__global__ void k_final(const float* __restrict__ alpha_p,
                        const float* __restrict__ beta_p,
                        float* __restrict__ ws,
                        float* __restrict__ out) {
    const int NT = 256, CH = T_LEN / NT;   // 32
    __shared__ float cs_e[NT], cs_w[NT], cs_v[NT];
    __shared__ float sc_e[NT], sc_w[NT], sc_v[NT];
    int t = threadIdx.x;
    const float* e = ws + WS_E;
    const float* v = ws + WS_V;
    float* Se  = ws + WS_SE;
    float* Sw  = ws + WS_SW;
    float* Sev = ws + WS_SEV;

    // per-chunk sums
    int base = t * CH;
    float se = 0.f, sw = 0.f, sv = 0.f;
    for (int i = 0; i < CH; ++i) {
        float ei = e[base + i];
        se += ei;
        sw += ei * (float)(T_LEN - 1 - (base + i));
        sv += ei * v[base + i];
    }
    cs_e[t] = se; cs_w[t] = sw; cs_v[t] = sv;
    __syncthreads();

    // exclusive suffix scan of chunk sums (serial on thread 0: 256 steps)
    if (t == 0) {
        float re = 0.f, rw = 0.f, rv = 0.f;
        for (int cidx = NT - 1; cidx >= 0; --cidx) {
            sc_e[cidx] = re; sc_w[cidx] = rw; sc_v[cidx] = rv;
            re += cs_e[cidx]; rw += cs_w[cidx]; rv += cs_v[cidx];
        }
    }
    __syncthreads();

    // per-element suffix arrays
    {
        float re = sc_e[t], rw = sc_w[t], rv = sc_v[t];
        for (int i = CH - 1; i >= 0; --i) {
            int j = base + i;
            float ei = e[j];
            re += ei;
            rw += ei * (float)(T_LEN - 1 - j);
            rv += ei * v[j];
            Se[j] = re; Sw[j] = rw; Sev[j] = rv;
        }
        if (t == NT - 1) { Se[T_LEN] = 0.f; Sw[T_LEN] = 0.f; Sev[T_LEN] = 0.f; }
    }
    __syncthreads();

    // sequential 30-iteration state machine (exact reference semantics)
    if (t == 0) {
        const float Tf = (float)T_LEN;
        float alpha = alpha_p[0], beta = beta_p[0];
        float m_old = 0.f, w_max = 0.f;
        bool  done = false;
        int   att_j = 0; float att_ssum = 1.f;
        for (int it = 0; it < N_ITERS; ++it) {
            float m        = 2.0f * (alpha + beta) / alpha;
            float window   = fminf(Tf, fabsf(ceilf(m)));
            float wmax_new = fmaxf(w_max, window);
            int jws = (int)(Tf - wmax_new);
            jws = jws < 0 ? 0 : (jws > T_LEN ? T_LEN : jws);
            float ssum = Se[jws];
            int jw = (int)(Tf - window);
            jw = jw < 0 ? 0 : (jw > T_LEN ? T_LEN : jw);
            float bupd = Sw[jw] / ssum;
            bool  brk  = (m > Tf) || (m < m_old);
            float alpha_new = alpha + 1.0f;
            float beta_new  = beta + bupd;
            float mold_new  = brk ? m_old : m;
            if (!done) {
                alpha = alpha_new; beta = beta_new; m_old = mold_new;
                w_max = wmax_new;  att_j = jw;      att_ssum = ssum;
            }
            done = done || brk;
        }
        out[0] = Sev[att_j] / att_ssum;   // y = sum(att * v_vec)
    }
}

// ------------------------------------------------------------------
extern "C" void kernel_launch(void* const* d_in, const int* in_sizes, int n_in,
                              void* d_out, int out_size, void* d_ws, size_t ws_size,
                              hipStream_t stream) {
    const float* x     = (const float*)d_in[0];
    const float* W     = (const float*)d_in[1];
    const float* alpha = (const float*)d_in[2];
    const float* beta  = (const float*)d_in[3];
    float* out = (float*)d_out;
    float* ws  = (float*)d_ws;

    k_qlast<<<N_EMBD / 256, 256, 0, stream>>>(x, W, ws);
    k_w2   <<<N_EMBD / 8,  256, 0, stream>>>(W, ws);          // 8 waves/block, wave per row
    k_ev   <<<(T_LEN / 16) / 8, 256, 0, stream>>>(x, ws);     // 8 waves/block, wave per tile
    k_final<<<1, 256, 0, stream>>>(alpha, beta, ws, out);
}